// MILoss_43009802502463
// MI455X (gfx1250) — compile-verified
//
#include <hip/hip_runtime.h>

typedef __attribute__((ext_vector_type(2))) float v2f;
typedef __attribute__((ext_vector_type(8))) float v8f;

#define MI_SAFE 1e-20f

#define WMMA_F32X4(A, B, C) \
    __builtin_amdgcn_wmma_f32_16x16x4_f32(false, (A), false, (B), (short)0, (C), false, false)

// ---------------------------------------------------------------------------
// jointN[s,h] = sum_i target[i,s]*input[i,h]  (8x8, K=N) via
// V_WMMA_F32_16X16X4_F32, packing TWO 4-row sample blocks per WMMA:
//   A rows 0-7  = targetT of rows [g*8,   g*8+4)   (block P)
//   A rows 8-15 = targetT of rows [g*8+4, g*8+8)   (block Q)
//   B cols 0-7  = input  of block P, cols 8-15 = input of block Q
// C[0:8,0:8] and C[8:16,8:16] are the two valid joint partials; the
// cross-quadrant garbage is never read. One 8-row group = 64 elements
// = exactly the footprint of its 4 loads (x/y for input/target): no
// duplicated or out-of-range bytes, so all full groups go unmasked.
//
// Operand layout (ISA 7.12.2): A 16x4: lane%16=M, lane/16 picks K pair,
// vector elt picks K in pair. B 4x16: v0 lo=K0 hi=K2, v1 lo=K1 hi=K3.
// Per-lane element offset within a group: (m>>3)*32 + kh*16 + (m&7), +8 for y.
// ---------------------------------------------------------------------------
__global__ __launch_bounds__(256)
void mi_joint_kernel(const float* __restrict__ input,
                     const float* __restrict__ target,
                     float* __restrict__ partials,   // [gridDim.x * 64]
                     long long nRows, long long nFast)   // nFast = nRows/8
{
    const int lane = threadIdx.x & 31;
    const int wib  = __builtin_amdgcn_readfirstlane(threadIdx.x >> 5); // scalar
    const int gwave  = (int)blockIdx.x * 8 + wib;
    const int nWaves = (int)gridDim.x * 8;

    const int m  = lane & 15;
    const int kh = lane >> 4;
    const int laneOff = ((m >> 3) << 5) + (kh << 4) + (m & 7);

    // Contiguous chunk of 8-row groups per wave (scalar bounds).
    long long chunk = (nFast + nWaves - 1) / nWaves;
    long long b0 = (long long)gwave * chunk;
    long long b1 = b0 + chunk; if (b1 > nFast) b1 = nFast;
    if (b0 > b1) b0 = b1;

    v8f c0 = {0.f,0.f,0.f,0.f,0.f,0.f,0.f,0.f};
    v8f c1 = {0.f,0.f,0.f,0.f,0.f,0.f,0.f,0.f};

    const float* pIn = input  + b0 * 64 + laneOff;
    const float* pTg = target + b0 * 64 + laneOff;

#define LOAD_GRP(j, a, b)                              \
    b.x = pIn[(j) * 64];  b.y = pIn[(j) * 64 + 8];     \
    a.x = pTg[(j) * 64];  a.y = pTg[(j) * 64 + 8];

#define BODY4()                                        \
    {                                                  \
        v2f aA, bA, aB, bB, aC, bC, aD, bD;            \
        LOAD_GRP(0, aA, bA)                            \
        LOAD_GRP(1, aB, bB)                            \
        LOAD_GRP(2, aC, bC)                            \
        LOAD_GRP(3, aD, bD)                            \
        c0 = WMMA_F32X4(aA, bA, c0);                   \
        c1 = WMMA_F32X4(aB, bB, c1);                   \
        c0 = WMMA_F32X4(aC, bC, c0);                   \
        c1 = WMMA_F32X4(aD, bD, c1);                   \
        pIn += 256; pTg += 256;                        \
    }

    long long blk = b0;
    long long bPF = (b1 - b0 > 16) ? (b1 - 16) : b0;   // prefetch-safe end
    // main loop with in-chunk prefetch (+1024 elements = 16 groups ahead)
    for (; blk + 4 <= bPF; blk += 4) {
        __builtin_prefetch(pIn + 1024, 0, 3);          // global_prefetch_b8
        __builtin_prefetch(pTg + 1024, 0, 3);
        BODY4();
    }
    // drain without prefetch
    for (; blk + 4 <= b1; blk += 4) {
        BODY4();
    }
    // remainder groups, one WMMA each, still unmasked
    for (; blk < b1; ++blk) {
        v2f a, b;
        b.x = pIn[0]; b.y = pIn[8];
        a.x = pTg[0]; a.y = pTg[8];
        c0 = WMMA_F32X4(a, b, c0);
        pIn += 64; pTg += 64;
    }
#undef BODY4
#undef LOAD_GRP

    // masked tail rows [nFast*8, nRows): wave 0 only, one group suffices
    if (gwave == 0 && nFast * 8 < nRows) {
        long long r = nFast * 8 + ((m >> 3) << 2) + 2 * kh;
        long long e = r * 8 + (m & 7);
        bool ok0 = (r     < nRows);
        bool ok1 = (r + 1 < nRows);
        v2f a, b;
        b.x = ok0 ? input [e    ] : 0.f;
        b.y = ok1 ? input [e + 8] : 0.f;
        a.x = ok0 ? target[e    ] : 0.f;
        a.y = ok1 ? target[e + 8] : 0.f;
        c0 = WMMA_F32X4(a, b, c0);
    }

    // merge accumulator chains
    #pragma unroll
    for (int v = 0; v < 8; ++v) c0[v] += c1[v];

    // Block P corner: lanes 0-7  (M=v,   N=lane).
    // Block Q corner: lanes 24-31 (M=v+8, N=lane-16); s=v, h=lane-24.
    __shared__ float red[8 * 2 * 64];
    if (lane < 8) {
        #pragma unroll
        for (int v = 0; v < 8; ++v)
            red[(wib * 2 + 0) * 64 + v * 8 + lane] = c0[v];
    } else if (lane >= 24) {
        #pragma unroll
        for (int v = 0; v < 8; ++v)
            red[(wib * 2 + 1) * 64 + v * 8 + (lane - 24)] = c0[v];
    }
    __syncthreads();

    int t = threadIdx.x;
    if (t < 64) {
        float s = 0.f;
        #pragma unroll
        for (int w = 0; w < 16; ++w) s += red[w * 64 + t];
        partials[(long long)blockIdx.x * 64 + t] = s;
    }
}

// ---------------------------------------------------------------------------
// Deterministic final reduction + MI epilogue (64 cells).
//   p_s_hat = colsum(jointN)/N  (exact: one-hot rows sum to 1)
//   p_s     = rowsum(jointN)/N  (= count[s]/N up to softmax rowsum rounding)
//   present = rowsum > 0
// ---------------------------------------------------------------------------
__global__ __launch_bounds__(64)
void mi_finalize_kernel(const float* __restrict__ partials, int numParts,
                        float* __restrict__ out, float invN)
{
    __shared__ float jointN[64];
    int t = threadIdx.x;
    if (t < 64) {
        float s = 0.f;
        for (int p = 0; p < numParts; ++p) s += partials[(long long)p * 64 + t];
        jointN[t] = s;
    }
    __syncthreads();
    if (t == 0) {
        float ps[8], psh[8];
        for (int i = 0; i < 8; ++i) { ps[i] = 0.f; psh[i] = 0.f; }
        for (int s = 0; s < 8; ++s)
            for (int h = 0; h < 8; ++h) {
                float j = jointN[s * 8 + h];
                ps[s]  += j;
                psh[h] += j;
            }
        bool present[8];
        for (int s = 0; s < 8; ++s) present[s] = (ps[s] > 0.f);

        float total = 0.f;
        for (int s = 0; s < 8; ++s) {
            float p_s = ps[s] * invN;
            if (p_s == 0.f) p_s = MI_SAFE;
            for (int h = 0; h < 8; ++h) {
                if (!(present[s] && present[h])) continue;
                float p_h = psh[h] * invN;
                if (p_h == 0.f) p_h = MI_SAFE;
                float j = jointN[s * 8 + h] * invN;
                if (j == 0.f) j = MI_SAFE;
                total += j * logf(j / (p_s * p_h));
            }
        }
        out[0] = total;
    }
}

extern "C" void kernel_launch(void* const* d_in, const int* in_sizes, int n_in,
                              void* d_out, int out_size, void* d_ws, size_t ws_size,
                              hipStream_t stream)
{
    const float* input_ = (const float*)d_in[0];   // [N, 8] f32 softmax probs
    const float* target = (const float*)d_in[1];   // [N, 8] f32 one-hot
    float* out = (float*)d_out;                    // scalar f32

    long long nTot  = (long long)in_sizes[0];
    long long nRows = nTot / 8;
    long long nFast = nRows / 8;                   // full 8-row groups

    int nBlocks = 1024;
    int maxBlocks = (int)(ws_size / (64 * sizeof(float)));
    if (nBlocks > maxBlocks) nBlocks = maxBlocks;
    if (nBlocks < 1) nBlocks = 1;
    long long maxUseful = (nFast + 7) / 8;         // >=1 group per wave
    if (maxUseful < 1) maxUseful = 1;
    if ((long long)nBlocks > maxUseful) nBlocks = (int)maxUseful;

    float* partials = (float*)d_ws;

    mi_joint_kernel<<<nBlocks, 256, 0, stream>>>(input_, target, partials,
                                                 nRows, nFast);
    mi_finalize_kernel<<<1, 64, 0, stream>>>(partials, nBlocks, out,
                                             1.0f / (float)nRows);
}